// Model_67568425500961
// MI455X (gfx1250) — compile-verified
//
#include <hip/hip_runtime.h>
#include <math.h>

// Problem constants (from reference)
#define DIM     128
#define NNODES  100000
#define NEDGES  1600000
#define NPAIRS  8192

#define MIN_NORM  1e-15f
#define MAX_TANH  15.0f
#define PROJ_EPS  0.004f
// c = 1.0 -> sqrt_c = 1.0, maxnorm = (1 - PROJ_EPS)

typedef float v2f __attribute__((ext_vector_type(2)));
typedef float v8f __attribute__((ext_vector_type(8)));

// ---------------- wave32 helpers ----------------
__device__ __forceinline__ float wave_sum(float v) {
#pragma unroll
  for (int o = 16; o > 0; o >>= 1) v += __shfl_xor(v, o, 32);
  return v;
}
__device__ __forceinline__ float wave_max(float v) {
#pragma unroll
  for (int o = 16; o > 0; o >>= 1) v = fmaxf(v, __shfl_xor(v, o, 32));
  return v;
}

__device__ __forceinline__ float4 f4scale(float4 v, float s) {
  return make_float4(v.x * s, v.y * s, v.z * s, v.w * s);
}
__device__ __forceinline__ float f4dot(float4 a, float4 b) {
  return a.x * b.x + a.y * b.y + a.z * b.z + a.w * b.w;
}

__device__ __forceinline__ float artanh_f(float x) {
  x = fminf(fmaxf(x, -1.0f + 1e-7f), 1.0f - 1e-7f);
  return 0.5f * (log1pf(x) - log1pf(-x));
}

// norm of a 128-float row distributed as float4 per lane (clipped)
__device__ __forceinline__ float row_norm(float4 v) {
  float s = wave_sum(f4dot(v, v));
  return fmaxf(sqrtf(s), MIN_NORM);
}

__device__ __forceinline__ float4 proj4(float4 v) {
  float n = row_norm(v);
  const float maxn = 1.0f - PROJ_EPS;
  float sc = (n > maxn) ? (maxn / n) : 1.0f;
  return f4scale(v, sc);
}

__device__ __forceinline__ float4 expmap0_4(float4 u) {
  float n = row_norm(u);                       // n >= MIN_NORM > 0
  float f = tanhf(fminf(n, MAX_TANH)) / n;     // sqrt_c = 1
  return f4scale(u, f);
}

__device__ __forceinline__ float4 logmap0_4(float4 p) {
  float n = row_norm(p);
  return f4scale(p, artanh_f(n) / n);
}

// ---------------- kernels ----------------

// hyp_b = proj(expmap0(b)) for both layers. 2 waves, one per bias.
__global__ void __launch_bounds__(64) k_hyp_bias(const float* __restrict__ b1,
                                                 const float* __restrict__ b2,
                                                 float* __restrict__ hb1,
                                                 float* __restrict__ hb2) {
  int wave = threadIdx.x >> 5, lane = threadIdx.x & 31;
  const float* b = wave ? b2 : b1;
  float* o = wave ? hb2 : hb1;
  float4 u = ((const float4*)b)[lane];
  float4 h = proj4(expmap0_4(u));
  ((float4*)o)[lane] = h;
}

// out = proj(expmap0(x)), row-wise; one wave per row.
__global__ void __launch_bounds__(256) k_expmap0_proj(const float* __restrict__ x,
                                                      float* __restrict__ out,
                                                      int nrows) {
  int wave = threadIdx.x >> 5, lane = threadIdx.x & 31;
  int row = blockIdx.x * 8 + wave;
  if (row >= nrows) return;
  float4 u = ((const float4*)(x + (size_t)row * DIM))[lane];
  float4 h = proj4(expmap0_4(u));
  ((float4*)(out + (size_t)row * DIM))[lane] = h;
}

// mv = h_in @ W via V_WMMA_F32_16X16X4_F32.
// W staged in LDS in K-pair-interleaved form: Wlds[p][c] = {W[2p][c], W[2p+1][c]}
// so every B fragment is ONE aligned ds_load_b64 into an even VGPR pair.
// Each wave computes a 32x128 output strip (two 16-row tiles x 8 col tiles);
// each B fragment feeds two WMMAs -> half the DS traffic per matrix op.
__global__ void __launch_bounds__(256) k_gemm_wmma(const float* __restrict__ A,
                                                   const float* __restrict__ W,
                                                   float* __restrict__ Cout,
                                                   int nrows) {
  __shared__ float2 Wlds[(DIM / 2) * DIM];  // 64 x 128 float2 = 64 KB
  int tid = threadIdx.x;

  // Stage W with pair interleave: unit u -> pair-row p, even col c2.
#pragma unroll
  for (int i = 0; i < 16; ++i) {
    int u = tid + i * 256;        // 0..4095
    int p = u >> 6;               // 0..63
    int c2 = (u & 63) * 2;        // even column
    float2 r0 = *(const float2*)(W + (size_t)(2 * p) * DIM + c2);
    float2 r1 = *(const float2*)(W + (size_t)(2 * p + 1) * DIM + c2);
    Wlds[p * DIM + c2]     = make_float2(r0.x, r1.x);
    Wlds[p * DIM + c2 + 1] = make_float2(r0.y, r1.y);
  }
  __syncthreads();

  int wave = tid >> 5, lane = tid & 31;
  int tile = blockIdx.x * 8 + wave;
  int row0 = tile * 32;
  if (row0 >= nrows) return;  // wave-uniform exit: WMMA waves keep full EXEC

  int kb   = (lane >> 4) * 2;         // A/B-frag K base per half-wave
  int half = lane >> 4;               // 0 or 1
  int col  = lane & 15;               // B/D-frag: N = lane&15

  v8f acc0[8], acc1[8];
#pragma unroll
  for (int t = 0; t < 8; ++t) { acc0[t] = (v8f){}; acc1[t] = (v8f){}; }

  const float* Arow0 = A + (size_t)(row0 + (lane & 15)) * DIM;
  const float* Arow1 = Arow0 + 16 * DIM;

  for (int kk = 0; kk < DIM; kk += 4) {
    float2 av0 = *(const float2*)(Arow0 + kk + kb);
    float2 av1 = *(const float2*)(Arow1 + kk + kb);
    v2f a0; a0.x = av0.x; a0.y = av0.y;
    v2f a1; a1.x = av1.x; a1.y = av1.y;
    const float2* Wrow = Wlds + ((kk >> 1) + half) * DIM;
#pragma unroll
    for (int t = 0; t < 8; ++t) {
      float2 bv = Wrow[t * 16 + col];    // one ds_load_b64
      v2f b; b.x = bv.x; b.y = bv.y;
      acc0[t] = __builtin_amdgcn_wmma_f32_16x16x4_f32(
          false, a0, false, b, (short)0, acc0[t], false, false);
      acc1[t] = __builtin_amdgcn_wmma_f32_16x16x4_f32(
          false, a1, false, b, (short)0, acc1[t], false, false);
    }
  }

  int rofs = half * 8;  // D layout: lanes16-31 hold M = v + 8
#pragma unroll
  for (int t = 0; t < 8; ++t) {
#pragma unroll
    for (int v = 0; v < 8; ++v) {
      Cout[(size_t)(row0 + rofs + v) * DIM + t * 16 + col]      = acc0[t][v];
      Cout[(size_t)(row0 + 16 + rofs + v) * DIM + t * 16 + col] = acc1[t][v];
    }
  }
}

// ht = logmap0(proj(mobius_add(proj(mobius_matvec_scale(mv, h_in)), hyp_b)))
__global__ void __launch_bounds__(256) k_post_matvec(const float* __restrict__ mv,
                                                     const float* __restrict__ hin,
                                                     const float* __restrict__ hb,
                                                     float* __restrict__ ht,
                                                     int nrows) {
  int wave = threadIdx.x >> 5, lane = threadIdx.x & 31;
  int row = blockIdx.x * 8 + wave;
  if (row >= nrows) return;
  float4 mx = ((const float4*)(mv + (size_t)row * DIM))[lane];
  float4 x  = ((const float4*)(hin + (size_t)row * DIM))[lane];

  float x_norm  = row_norm(x);
  float mx_norm = row_norm(mx);
  float amax = wave_max(fmaxf(fmaxf(fabsf(mx.x), fabsf(mx.y)),
                              fmaxf(fabsf(mx.z), fabsf(mx.w))));
  float arg = fminf(fmaxf(mx_norm / x_norm * artanh_f(x_norm), -MAX_TANH), MAX_TANH);
  float f = tanhf(arg) / mx_norm;  // sqrt_c = 1
  float4 res = f4scale(mx, f);
  if (amax == 0.0f) res = make_float4(0.f, 0.f, 0.f, 0.f);  // zero_rows
  res = proj4(res);

  float4 y = ((const float4*)hb)[lane];
  // mobius_add(res, y), c = 1
  float x2 = wave_sum(f4dot(res, res));
  float y2 = wave_sum(f4dot(y, y));
  float xy = wave_sum(f4dot(res, y));
  float ca  = 1.0f + 2.0f * xy + y2;
  float cb  = 1.0f - x2;
  float den = fmaxf(1.0f + 2.0f * xy + x2 * y2, MIN_NORM);
  float inv = 1.0f / den;
  float4 hm = make_float4((ca * res.x + cb * y.x) * inv,
                          (ca * res.y + cb * y.y) * inv,
                          (ca * res.z + cb * y.z) * inv,
                          (ca * res.w + cb * y.w) * inv);
  hm = proj4(hm);
  float4 lt = logmap0_4(hm);
  ((float4*)(ht + (size_t)row * DIM))[lane] = lt;
}

__global__ void __launch_bounds__(256) k_zero(float4* __restrict__ p, size_t n4) {
  size_t i = blockIdx.x * (size_t)blockDim.x + threadIdx.x;
  size_t stride = gridDim.x * (size_t)blockDim.x;
  float4 z = make_float4(0.f, 0.f, 0.f, 0.f);
  for (; i < n4; i += stride) p[i] = z;
}

// agg[dst] += w * ht[src]; one wave per edge, float4 per lane, L2 atomics.
// ht (51 MB) and agg (51 MB) both fit in the 192 MB L2, so gathers and
// float atomics stay on-chip; HBM only sees the edge lists.
__global__ void __launch_bounds__(256) k_scatter(const float* __restrict__ ht,
                                                 const int* __restrict__ src,
                                                 const int* __restrict__ dst,
                                                 const float* __restrict__ ew,
                                                 float* __restrict__ agg,
                                                 int nedges) {
  int wave = threadIdx.x >> 5, lane = threadIdx.x & 31;
  int e = blockIdx.x * 8 + wave;
  if (e >= nedges) return;
  int s = src[e], d = dst[e];
  float w = ew[e];
  float4 m = ((const float4*)(ht + (size_t)s * DIM))[lane];
  float* a = agg + (size_t)d * DIM + lane * 4;
  atomicAdd(a + 0, w * m.x);
  atomicAdd(a + 1, w * m.y);
  atomicAdd(a + 2, w * m.z);
  atomicAdd(a + 3, w * m.w);
}

// h_out = proj(expmap0(relu(logmap0(proj(expmap0(agg))))))
__global__ void __launch_bounds__(256) k_post_agg(const float* __restrict__ agg,
                                                  float* __restrict__ hout,
                                                  int nrows) {
  int wave = threadIdx.x >> 5, lane = threadIdx.x & 31;
  int row = blockIdx.x * 8 + wave;
  if (row >= nrows) return;
  float4 a = ((const float4*)(agg + (size_t)row * DIM))[lane];
  float4 h = proj4(expmap0_4(a));
  float4 u = logmap0_4(h);
  u.x = fmaxf(u.x, 0.f); u.y = fmaxf(u.y, 0.f);
  u.z = fmaxf(u.z, 0.f); u.w = fmaxf(u.w, 0.f);
  float4 o = proj4(expmap0_4(u));
  ((float4*)(hout + (size_t)row * DIM))[lane] = o;
}

// Fermi-Dirac decoder over hcat = [h1 | h2]; one wave per pair.
__global__ void __launch_bounds__(256) k_pairs(const float* __restrict__ h1,
                                               const float* __restrict__ h2,
                                               const int* __restrict__ idx,
                                               float* __restrict__ out,
                                               int npairs) {
  int wave = threadIdx.x >> 5, lane = threadIdx.x & 31;
  int p = blockIdx.x * 8 + wave;
  if (p >= npairs) return;
  int i = idx[2 * p + 0];
  int j = idx[2 * p + 1];
  float4 a1 = ((const float4*)(h1 + (size_t)i * DIM))[lane];
  float4 b1 = ((const float4*)(h1 + (size_t)j * DIM))[lane];
  float4 a2 = ((const float4*)(h2 + (size_t)i * DIM))[lane];
  float4 b2 = ((const float4*)(h2 + (size_t)j * DIM))[lane];
  float4 d1 = make_float4(a1.x - b1.x, a1.y - b1.y, a1.z - b1.z, a1.w - b1.w);
  float4 d2 = make_float4(a2.x - b2.x, a2.y - b2.y, a2.z - b2.z, a2.w - b2.w);
  float s = wave_sum(f4dot(d1, d1) + f4dot(d2, d2));
  if (lane == 0) {
    float dist = sqrtf(fmaxf(s, MIN_NORM));
    out[p] = 1.0f / (expf((dist - 2.0f) / 1.0f) + 1.0f);  // FD_R=2, FD_T=1
  }
}

// ---------------- host launcher ----------------
extern "C" void kernel_launch(void* const* d_in, const int* in_sizes, int n_in,
                              void* d_out, int out_size, void* d_ws, size_t ws_size,
                              hipStream_t stream) {
  const float* x   = (const float*)d_in[0];
  const float* ew  = (const float*)d_in[1];
  const float* W1  = (const float*)d_in[2];
  const float* b1  = (const float*)d_in[3];
  const float* W2  = (const float*)d_in[4];
  const float* b2  = (const float*)d_in[5];
  const int* esrc  = (const int*)d_in[6];
  const int* edst  = (const int*)d_in[7];
  const int* idx   = (const int*)d_in[8];
  float* out = (float*)d_out;

  const size_t nd = (size_t)NNODES * DIM;
  float* hb1 = (float*)d_ws;
  float* hb2 = hb1 + DIM;
  float* B0  = hb2 + DIM;        // h0 (then h2)
  float* B1  = B0 + nd;          // mv
  float* B2  = B1 + nd;          // ht
  float* B3  = B2 + nd;          // agg
  float* B4  = B3 + nd;          // h1

  const int rowBlocks  = (NNODES + 7) / 8;           // 12500
  const int tileBlocks = ((NNODES / 32) + 7) / 8;    // 391 (N % 32 == 0)
  const int edgeBlocks = (NEDGES + 7) / 8;           // 200000
  const int pairBlocks = (NPAIRS + 7) / 8;           // 1024

  k_hyp_bias<<<1, 64, 0, stream>>>(b1, b2, hb1, hb2);
  k_expmap0_proj<<<rowBlocks, 256, 0, stream>>>(x, B0, NNODES);

  // ---- layer 1 ----
  k_gemm_wmma<<<tileBlocks, 256, 0, stream>>>(B0, W1, B1, NNODES);
  k_post_matvec<<<rowBlocks, 256, 0, stream>>>(B1, B0, hb1, B2, NNODES);
  k_zero<<<2048, 256, 0, stream>>>((float4*)B3, nd / 4);
  k_scatter<<<edgeBlocks, 256, 0, stream>>>(B2, esrc, edst, ew, B3, NEDGES);
  k_post_agg<<<rowBlocks, 256, 0, stream>>>(B3, B4, NNODES);  // B4 = h1

  // ---- layer 2 ----
  k_gemm_wmma<<<tileBlocks, 256, 0, stream>>>(B4, W2, B1, NNODES);
  k_post_matvec<<<rowBlocks, 256, 0, stream>>>(B1, B4, hb2, B2, NNODES);
  k_zero<<<2048, 256, 0, stream>>>((float4*)B3, nd / 4);
  k_scatter<<<edgeBlocks, 256, 0, stream>>>(B2, esrc, edst, ew, B3, NEDGES);
  k_post_agg<<<rowBlocks, 256, 0, stream>>>(B3, B0, NNODES);  // B0 = h2

  // ---- decoder ----
  k_pairs<<<pairBlocks, 256, 0, stream>>>(B4, B0, idx, out, NPAIRS);
}